// LoadPathLoss_8942121910601
// MI455X (gfx1250) — compile-verified
//
#include <hip/hip_runtime.h>
#include <stdint.h>

// Problem constants (match reference setup_inputs)
#define BB 4
#define CC 8
#define DD 64
#define HH 128
#define WW 128
#define PLANE (HH * WW)          // 16384 pixels = 64 KiB per z-slice
#define NTHREADS 1024            // 32 wave32s
#define PPT (PLANE / NTHREADS)   // 16 pixels per thread (contiguous, one row segment)

// s_wait_asynccnt with immediate; asm keeps it a compiler memory barrier too.
#define WAIT_ASYNC(n) asm volatile("s_wait_asynccnt %0" ::"i"(n) : "memory")

// Issue 4x b128 async global->LDS loads covering this thread's 64B chunk of a
// 64KiB z-slice. GVS mode: saddr = slice base (SGPR pair), vaddr = byte offset.
__device__ __forceinline__ void async_load_slice(unsigned lds_base_b,
                                                 const float* gbase,
                                                 unsigned toff_b) {
  unsigned long long sa = (unsigned long long)(uintptr_t)gbase;
#pragma unroll
  for (int k = 0; k < 4; ++k) {
    unsigned dst = lds_base_b + toff_b + (unsigned)(k * 16);
    unsigned off = toff_b + (unsigned)(k * 16);
    asm volatile("global_load_async_to_lds_b128 %0, %1, %2"
                 :
                 : "v"(dst), "v"(off), "s"(sa)
                 : "memory");
  }
}

extern "C" __global__ __launch_bounds__(NTHREADS, 1) void support_scan_kernel(
    const float* __restrict__ state, const int* __restrict__ ch_ptr,
    float* __restrict__ partial) {
  extern __shared__ float smem[];
  float* sup = smem;                 // 16384 floats: supported plane
  float* buf0 = smem + PLANE;        // slice double-buffer A
  float* buf1 = smem + 2 * PLANE;    // slice double-buffer B
  float* wsum = smem + 3 * PLANE;    // 32 wave partials

  const int tid = threadIdx.x;
  const int b = blockIdx.x;
  const int ch = ch_ptr[0];
  // structure[b] base: state[b, ch, 0, 0, 0]
  const float* base = state + (size_t)((b * CC + ch)) * DD * PLANE;

  // Raw LDS byte addresses (low 32 bits of the flat LDS-aperture pointer).
  const unsigned sup_b = (unsigned)(uintptr_t)(void*)sup;
  const unsigned buf0_b = (unsigned)(uintptr_t)(void*)buf0;
  const unsigned buf1_b = (unsigned)(uintptr_t)(void*)buf1;
  const unsigned toff_b = (unsigned)(tid * 64);  // this thread's 64B chunk

  // Kick off: z=0 -> sup, z=1 -> buf1 (each wave: 8 outstanding async ops).
  async_load_slice(sup_b, base, toff_b);
  async_load_slice(buf1_b, base + (size_t)1 * PLANE, toff_b);
  WAIT_ASYNC(4);    // first 4 (sup) complete — async loads retire in order
  __syncthreads();  // every wave finished its share: sup plane fully valid

  float newv[PPT];

  for (int z = 1; z < DD; ++z) {
    // Prefetch slice z+1 into the other buffer while we compute slice z.
    if (z + 1 < DD) {
      async_load_slice(((z + 1) & 1) ? buf1_b : buf0_b,
                       base + (size_t)(z + 1) * PLANE, toff_b);
      WAIT_ASYNC(4);  // slice z (older 4 ops) has landed
    } else {
      WAIT_ASYNC(0);
    }
    const float* cur = (z & 1) ? buf1 : buf0;

    // This thread owns 16 consecutive pixels of one row: y fixed, x0..x0+15.
#pragma unroll
    for (int k = 0; k < PPT; ++k) {
      const int p = tid * PPT + k;
      const int y = p >> 7;
      const int x = p & (WW - 1);
      const float s = sup[p];
      float m = s;  // 3x3 window always contains the center; -inf pad = skip OOB
      const bool ym = (y > 0), yp = (y < HH - 1);
      const bool xm = (x > 0), xp = (x < WW - 1);
      if (ym) {
        m = fmaxf(m, sup[p - WW]);
        if (xm) m = fmaxf(m, sup[p - WW - 1]);
        if (xp) m = fmaxf(m, sup[p - WW + 1]);
      }
      if (yp) {
        m = fmaxf(m, sup[p + WW]);
        if (xm) m = fmaxf(m, sup[p + WW - 1]);
        if (xp) m = fmaxf(m, sup[p + WW + 1]);
      }
      if (xm) m = fmaxf(m, sup[p - 1]);
      if (xp) m = fmaxf(m, sup[p + 1]);
      const float st = cur[p];  // own 64B chunk — loaded by this thread's async ops
      newv[k] = fmaxf(s, fminf(st, m));
    }
    __syncthreads();  // all neighbor reads done before overwriting sup
#pragma unroll
    for (int k = 0; k < PPT; ++k) sup[tid * PPT + k] = newv[k];
    __syncthreads();  // new sup plane visible to all waves
  }

  // unsupported = relu(structure - sup_final), summed over all z and pixels.
  float acc = 0.0f;
  for (int z = 0; z < DD; ++z) {
    const float4* g = (const float4*)(base + (size_t)z * PLANE) + tid * 4;
#pragma unroll
    for (int k = 0; k < 4; ++k) {
      const float4 v = g[k];
      const int p = tid * PPT + k * 4;
      acc += fmaxf(v.x - sup[p + 0], 0.0f);
      acc += fmaxf(v.y - sup[p + 1], 0.0f);
      acc += fmaxf(v.z - sup[p + 2], 0.0f);
      acc += fmaxf(v.w - sup[p + 3], 0.0f);
    }
  }

  // wave32 tree reduce, then cross-wave via LDS.
#pragma unroll
  for (int off = 16; off > 0; off >>= 1) acc += __shfl_down(acc, off, 32);
  if ((tid & 31) == 0) wsum[tid >> 5] = acc;
  __syncthreads();
  if (tid == 0) {
    float s = 0.0f;
#pragma unroll
    for (int i = 0; i < NTHREADS / 32; ++i) s += wsum[i];
    partial[b] = s;  // plain overwrite: deterministic across replays
  }
}

extern "C" __global__ void finalize_kernel(const float* __restrict__ partial,
                                           float* __restrict__ out) {
  const float inv_n = 1.0f / (float)((size_t)BB * DD * PLANE);
  out[0] = (partial[0] + partial[1] + partial[2] + partial[3]) * inv_n;
}

extern "C" void kernel_launch(void* const* d_in, const int* in_sizes, int n_in,
                              void* d_out, int out_size, void* d_ws,
                              size_t ws_size, hipStream_t stream) {
  (void)in_sizes; (void)n_in; (void)out_size; (void)ws_size;
  const float* state = (const float*)d_in[0];
  const int* ch = (const int*)d_in[1];
  float* partial = (float*)d_ws;  // 4 floats of scratch, overwritten every call

  // Dynamic LDS: sup (64K) + two slice buffers (128K) + 32 wave partials.
  const size_t smem_bytes = (size_t)(3 * PLANE) * sizeof(float) + 32 * sizeof(float);
  support_scan_kernel<<<dim3(BB), dim3(NTHREADS), smem_bytes, stream>>>(state, ch,
                                                                        partial);
  finalize_kernel<<<1, 1, 0, stream>>>(partial, (float*)d_out);
}